// ChebyNet_3083786518792
// MI455X (gfx1250) — compile-verified
//
#include <hip/hip_runtime.h>
#include <cstddef>

// Problem constants (match the reference)
#define NN   10000
#define EE   320000
#define DIN  256
#define DHID 256
#define DOUT 128

typedef __attribute__((ext_vector_type(2))) float v2f;
typedef __attribute__((ext_vector_type(8))) float v8f;

// ---------------- small elementwise / scatter kernels ----------------

__global__ void fill_zero_kernel(float* __restrict__ p, int n) {
    int i = blockIdx.x * blockDim.x + threadIdx.x;
    if (i < n) p[i] = 0.0f;
}

__global__ void neg_copy_kernel(float* __restrict__ o, const float* __restrict__ a, int n) {
    int i = blockIdx.x * blockDim.x + threadIdx.x;
    if (i < n) o[i] = -a[i];
}

__global__ void deg_accum_kernel(const int* __restrict__ dst, float* __restrict__ deg, int e) {
    int i = blockIdx.x * blockDim.x + threadIdx.x;
    if (i < e) atomicAdd(&deg[dst[i]], 1.0f);
}

__global__ void deg_finalize_kernel(float* __restrict__ deg, int n) {
    int i = blockIdx.x * blockDim.x + threadIdx.x;
    if (i < n) {
        float d = deg[i];
        deg[i] = (d > 0.0f) ? rsqrtf(fmaxf(d, 1.0f)) : 0.0f;
    }
}

__global__ void norm_kernel(const int* __restrict__ src, const int* __restrict__ dst,
                            const float* __restrict__ dis, float* __restrict__ nrm, int e) {
    int i = blockIdx.x * blockDim.x + threadIdx.x;
    if (i < e) nrm[i] = -dis[src[i]] * dis[dst[i]];
}

// Sparse propagation: out[dst[e], :] += scale * norm[e] * t[src[e], :]
// 256 features per node; 64 threads handle one edge (float4 each) -> coalesced
// 1KB gather + scattered global_atomic_add_f32.
__global__ void prop256_kernel(const float* __restrict__ t,
                               const int* __restrict__ src, const int* __restrict__ dst,
                               const float* __restrict__ nrm, float scale,
                               float* __restrict__ out, int e_cnt) {
    int tid = blockIdx.x * blockDim.x + threadIdx.x;
    int e = tid >> 6;
    if (e >= e_cnt) return;
    int c = (tid & 63) << 2;
    int s = src[e];
    int d = dst[e];
    float w = nrm[e] * scale;
    const float4 v = *reinterpret_cast<const float4*>(t + (size_t)s * 256 + c);
    float* o = out + (size_t)d * 256 + c;
    atomicAdd(o + 0, v.x * w);
    atomicAdd(o + 1, v.y * w);
    atomicAdd(o + 2, v.z * w);
    atomicAdd(o + 3, v.w * w);
}

// ---------------- fused Chebyshev triple-GEMM, f32 WMMA + LDS-staged B ----------------
// out[M, ODIM] = act( A0@W[0] + A1@W[1] + A2@W[2] + bias ), A* are [M, KDIM].
// Block = 8 waves; block owns one 16-col N-tile and 8 consecutive 16-row M-tiles.
// The transposed weight slab for all 3 Chebyshev terms (3 x 16 x KDIM f32, padded
// row stride) is staged in LDS once per block; each WMMA then needs only one
// global b64 (A frag) + one ds b64 (B frag).
//
// v_wmma_f32_16x16x4_f32 fragment layouts (wave32):
//   A (16x4): lane&15 -> M row; lanes<16 hold K{0,1} in {v0,v1}; lanes>=16 K{2,3}
//   B (4x16): lane&15 -> N col; lanes<16 hold K rows {0,1}; lanes>=16 rows {2,3}
//   C/D (16x16): lane&15 -> N col; vgpr v -> row v (+8 for lanes>=16)
template <int KDIM, int ODIM, bool RELU>
__global__ void cheb_gemm_kernel(const float* __restrict__ A0,
                                 const float* __restrict__ A1,
                                 const float* __restrict__ A2,
                                 const float* __restrict__ W,   // [3, KDIM, ODIM]
                                 const float* __restrict__ bias,
                                 float* __restrict__ out,       // [M, ODIM]
                                 int mtiles) {
    constexpr int KPAD = KDIM + 2;            // pad: lane j -> LDS bank 2j (no conflicts)
    __shared__ float wslab[3 * 16 * KPAD];    // ~48.4 KB for KDIM=256

    const int nt = blockIdx.x;                // N-tile of this block
    const int n0 = nt * 16;

    // ---- cooperative stage: wslab[(s*16+n)*KPAD + k] = W[s][k][n0+n] ----
    // flat idx = ((s*KDIM)+k)*16 + n  -> consecutive threads read consecutive n
    // (16-float contiguous segments in global memory).
    constexpr int TOT = 3 * KDIM * 16;
    for (int idx = threadIdx.x; idx < TOT; idx += blockDim.x) {
        const int n = idx & 15;
        const int k = (idx >> 4) & (KDIM - 1);
        const int s = idx >> 12;               // KDIM == 256
        wslab[(s * 16 + n) * KPAD + k] = W[((size_t)s * KDIM + k) * ODIM + n0 + n];
    }
    __syncthreads();

    const int lane = threadIdx.x & 31;
    const int wave = threadIdx.x >> 5;
    const int mt = blockIdx.y * 8 + wave;      // wave-uniform
    if (mt >= mtiles) return;                  // whole-wave exit after barrier: EXEC all-1s

    const int mrow = mt * 16 + (lane & 15);
    const int ncol = n0 + (lane & 15);
    const int koff = (lane < 16) ? 0 : 2;

    // bias depends only on the column -> same value in all 8 accumulator VGPRs
    const float bv = bias[ncol];
    v8f acc;
#pragma unroll
    for (int i = 0; i < 8; ++i) acc[i] = bv;

    const float* As[3] = {A0, A1, A2};
#pragma unroll
    for (int s = 0; s < 3; ++s) {
        const float* __restrict__ Ap = As[s] + (size_t)mrow * KDIM + koff;
        const float* __restrict__ Bp = &wslab[(s * 16 + (lane & 15)) * KPAD + koff];
#pragma unroll 8
        for (int k = 0; k < KDIM; k += 4) {
            v2f a;
            a.x = Ap[k];            // contiguous pair -> global b64 load
            a.y = Ap[k + 1];
            v2f b;
            b.x = Bp[k];            // contiguous pair -> ds_load_b64, conflict-free
            b.y = Bp[k + 1];
            acc = __builtin_amdgcn_wmma_f32_16x16x4_f32(
                /*neg_a=*/false, a, /*neg_b=*/false, b,
                /*c_mod=*/(short)0, acc, /*reuse_a=*/false, /*reuse_b=*/false);
        }
    }

    const int rowbase = mt * 16 + ((lane < 16) ? 0 : 8);
#pragma unroll
    for (int v = 0; v < 8; ++v) {
        float r = acc[v];
        if (RELU) r = fmaxf(r, 0.0f);
        out[(size_t)(rowbase + v) * ODIM + ncol] = r;
    }
}

// ---------------- host-side orchestration ----------------

extern "C" void kernel_launch(void* const* d_in, const int* in_sizes, int n_in,
                              void* d_out, int out_size, void* d_ws, size_t ws_size,
                              hipStream_t stream) {
    const float* x  = (const float*)d_in[0];          // [N, 256]
    const int*   ed = (const int*)d_in[1];            // [2, E]
    const float* W1 = (const float*)d_in[2];          // [3, 256, 256]
    const float* b1 = (const float*)d_in[3];          // [256]
    const float* W2 = (const float*)d_in[4];          // [3, 256, 128]
    const float* b2 = (const float*)d_in[5];          // [128]
    float* out = (float*)d_out;                       // [N, 128]

    const int* src = ed;
    const int* dst = ed + EE;

    // Workspace layout (floats). Total ~32 MB.
    float* ws   = (float*)d_ws;
    float* dis  = ws;                                  // [16384]   (deg, then rsqrt)
    float* nrm  = dis + 16384;                         // [E]
    float* bufA = nrm + EE;                            // [N*256]  Tx1
    float* bufB = bufA + (size_t)NN * 256;             // [N*256]  Tx2
    float* h    = bufB + (size_t)NN * 256;             // [N*256]  layer-1 output

    const int TB = 256;
    const int nfeat  = NN * 256;
    const int gN     = (NN + TB - 1) / TB;
    const int gE     = (EE + TB - 1) / TB;
    const int gFeat  = (nfeat + TB - 1) / TB;
    const int gProp  = (EE * 64 + TB - 1) / TB;        // 64 threads per edge
    const int mtiles = NN / 16;                        // 625 (exact)
    const int mblk   = (mtiles + 7) / 8;               // 79

    // ---- normalization ----
    fill_zero_kernel<<<gN, TB, 0, stream>>>(dis, NN);
    deg_accum_kernel<<<gE, TB, 0, stream>>>(dst, dis, EE);
    deg_finalize_kernel<<<gN, TB, 0, stream>>>(dis, NN);
    norm_kernel<<<gE, TB, 0, stream>>>(src, dst, dis, nrm, EE);

    // ---- layer 1: Tx0 = x, Tx1 = prop(x), Tx2 = 2*prop(Tx1) - x ----
    fill_zero_kernel<<<gFeat, TB, 0, stream>>>(bufA, nfeat);
    prop256_kernel<<<gProp, TB, 0, stream>>>(x, src, dst, nrm, 1.0f, bufA, EE);
    neg_copy_kernel<<<gFeat, TB, 0, stream>>>(bufB, x, nfeat);
    prop256_kernel<<<gProp, TB, 0, stream>>>(bufA, src, dst, nrm, 2.0f, bufB, EE);

    // h = relu(x@W1[0] + Tx1@W1[1] + Tx2@W1[2] + b1)
    {
        dim3 grid(DHID / 16, mblk);
        cheb_gemm_kernel<DIN, DHID, true><<<grid, TB, 0, stream>>>(x, bufA, bufB, W1, b1, h, mtiles);
    }

    // ---- layer 2 on h ----
    fill_zero_kernel<<<gFeat, TB, 0, stream>>>(bufA, nfeat);
    prop256_kernel<<<gProp, TB, 0, stream>>>(h, src, dst, nrm, 1.0f, bufA, EE);
    neg_copy_kernel<<<gFeat, TB, 0, stream>>>(bufB, h, nfeat);
    prop256_kernel<<<gProp, TB, 0, stream>>>(bufA, src, dst, nrm, 2.0f, bufB, EE);

    // out = h@W2[0] + Tx1@W2[1] + Tx2@W2[2] + b2
    {
        dim3 grid(DOUT / 16, mblk);
        cheb_gemm_kernel<DHID, DOUT, false><<<grid, TB, 0, stream>>>(h, bufA, bufB, W2, b2, out, mtiles);
    }
}